// AI4DEM_1357209666429
// MI455X (gfx1250) — compile-verified
//
#include <hip/hip_runtime.h>

// AI4DEM 5x5 contact-force stencil on a 2048x2048 torus, fp32.
// Tile staged to LDS via gfx1250 async global->LDS loads (ASYNCcnt),
// one v_rsq_f32 per neighbor channel (replaces sqrt + divide).

#define NYX   2048
#define WMASK 2047          // NYX is a power of two -> wrap via AND
#define TX    64            // output tile width
#define TY    16            // output tile height
#define PW    (TX + 4)      // padded (halo) tile width  = 68
#define PH    (TY + 4)      // padded (halo) tile height = 20
#define NTHREADS 256        // 8 wave32s

#define AS1 __attribute__((address_space(1)))
#define AS3 __attribute__((address_space(3)))

__device__ __forceinline__ float fast_rsqrt(float x) {
#if __has_builtin(__builtin_amdgcn_rsqf)
    return __builtin_amdgcn_rsqf(x);   // v_rsq_f32
#else
    return rsqrtf(x);
#endif
}

__global__ __launch_bounds__(NTHREADS)
void AI4DEM_force_kernel(const float* __restrict__ grid_x,
                         const float* __restrict__ grid_y,
                         const float* __restrict__ mask,
                         const int*   __restrict__ d_p,
                         const int*   __restrict__ kn_p,
                         float*       __restrict__ out)
{
    __shared__ float sgx[PH * PW];
    __shared__ float sgy[PH * PW];

    const int tid = threadIdx.x;
    const int bx  = blockIdx.x;           // tile column
    const int by  = blockIdx.y;           // tile row
    const int gy0 = by * TY - 2;          // halo origin (may be negative -> wrap)
    const int gx0 = bx * TX - 2;

    // ---- stage padded tile (torus wraparound) into LDS ----
#if __has_builtin(__builtin_amdgcn_global_load_async_to_lds_b32)
    for (int p = tid; p < PH * PW; p += NTHREADS) {
        const int pr   = p / PW;
        const int pc   = p - pr * PW;
        const int gy   = (gy0 + pr) & WMASK;
        const int gx   = (gx0 + pc) & WMASK;
        const int gidx = gy * NYX + gx;
        __builtin_amdgcn_global_load_async_to_lds_b32(
            (AS1 int*)(grid_x + gidx), (AS3 int*)(&sgx[p]), 0, 0);
        __builtin_amdgcn_global_load_async_to_lds_b32(
            (AS1 int*)(grid_y + gidx), (AS3 int*)(&sgy[p]), 0, 0);
    }
    asm volatile("s_wait_asynccnt 0x0" ::: "memory");
#else
    for (int p = tid; p < PH * PW; p += NTHREADS) {
        const int pr   = p / PW;
        const int pc   = p - pr * PW;
        const int gy   = (gy0 + pr) & WMASK;
        const int gx   = (gx0 + pc) & WMASK;
        const int gidx = gy * NYX + gx;
        sgx[p] = grid_x[gidx];
        sgy[p] = grid_y[gidx];
    }
#endif
    __syncthreads();

    // ---- scalars (uniform loads) ----
    const float knf     = (float)(*kn_p);
    const float dd      = (float)(*d_p);
    const float two_d   = 2.0f * dd;
    const float four_d2 = two_d * two_d;
    const float inv_eps = 1.0e4f;          // 1/EPS, EPS = 1e-4

    // ---- register-block: each thread does 4 vertically adjacent cells ----
    const int lx = tid & (TX - 1);         // 0..63
    const int qy = tid >> 6;               // 0..3
    const int r0 = 4 * qy;                 // first output row in tile

    float wx[8][5], wy[8][5];              // 8x5 window covering 4 cells' 5x5s
    const int base = r0 * PW + lx;
#pragma unroll
    for (int r = 0; r < 8; ++r) {
#pragma unroll
        for (int c = 0; c < 5; ++c) {
            wx[r][c] = sgx[base + r * PW + c];
            wy[r][c] = sgy[base + r * PW + c];
        }
    }

    float accx[4], accy[4];
#pragma unroll
    for (int k = 0; k < 4; ++k) {
        const float cx = wx[k + 2][2];     // padded center of output row r0+k
        const float cy = wy[k + 2][2];
        float fx = 0.0f, fy = 0.0f;
#pragma unroll
        for (int oy = 0; oy < 5; ++oy) {
#pragma unroll
            for (int ox = 0; ox < 5; ++ox) {
                const float ax = cx - wx[k + oy][ox];
                const float ay = cy - wy[k + oy][ox];
                const float sq = fmaf(ax, ax, ay * ay);
                // one TRANS op per channel:
                //   rs   = rsqrt(sq)            (clamped so sq==0 -> dist==0)
                //   dist = sq * rs  == sqrt(sq)
                //   inv  = min(rs, 1/EPS) == 1/max(EPS, dist)
                const float rs   = fast_rsqrt(fmaxf(sq, 1.0e-20f));
                const float dist = sq * rs;
                const float inv  = fminf(rs, inv_eps);
                const float coef = (knf * inv) * (dist - two_d);
                const bool  contact = sq < four_d2;   // dist < 2d
                fx += contact ? coef * ax : 0.0f;
                fy += contact ? coef * ay : 0.0f;
            }
        }
        accx[k] = fx;
        accy[k] = fy;
    }

    // ---- mask + coalesced store (fx plane, then fy plane) ----
    const int gxo = bx * TX + lx;
#pragma unroll
    for (int k = 0; k < 4; ++k) {
        const int   gyo = by * TY + r0 + k;
        const int   idx = gyo * NYX + gxo;
        const float m   = mask[idx];
        out[idx]             = accx[k] * m;
        out[NYX * NYX + idx] = accy[k] * m;
    }
}

extern "C" void kernel_launch(void* const* d_in, const int* in_sizes, int n_in,
                              void* d_out, int out_size, void* d_ws, size_t ws_size,
                              hipStream_t stream) {
    (void)in_sizes; (void)n_in; (void)out_size; (void)d_ws; (void)ws_size;
    const float* grid_x = (const float*)d_in[0];
    const float* grid_y = (const float*)d_in[1];
    const float* mask   = (const float*)d_in[2];
    const int*   d_p    = (const int*)d_in[3];
    const int*   kn_p   = (const int*)d_in[4];
    float*       out    = (float*)d_out;

    dim3 grid(NYX / TX, NYX / TY);   // 32 x 128 = 4096 workgroups
    AI4DEM_force_kernel<<<grid, NTHREADS, 0, stream>>>(
        grid_x, grid_y, mask, d_p, kn_p, out);
}